// CapsuleLayer_85693187490407
// MI455X (gfx1250) — compile-verified
//
#include <hip/hip_runtime.h>
#include <math.h>

#define IN_DIM 512
#define NCAPS  16
#define ODIM   32
#define MT     64          // batch rows per workgroup
#define KC     32          // K chunk staged in LDS
#define NCHUNK (IN_DIM / KC)
#define XPITCH 36          // padded x-chunk row stride (floats): 16B aligned, bank-conflict-free
#define WPITCH 528         // padded W-chunk row stride (floats): lane-halves hit disjoint banks
#define UPITCH 516         // padded routing-u row stride (floats)
#define EPSV   1e-8f

typedef float v2f __attribute__((ext_vector_type(2)));
typedef float v8f __attribute__((ext_vector_type(8)));

__global__ __launch_bounds__(512) void caps_fused_kernel(
    const float* __restrict__ x,   // [32768, 512]
    const float* __restrict__ W,   // [512, 512]
    float* __restrict__ out)       // [32768, 32]
{
    __shared__ float xs[2][MT * XPITCH];   // 2 x 9 KB   x chunk [64 rows][KC]
    __shared__ float ws[2][KC * WPITCH];   // 2 x 66 KB  W chunk [KC][512]

    const int tid  = threadIdx.x;
    const int wave = tid >> 5;
    const int lane = tid & 31;
    const int half = lane >> 4;
    const int ln   = lane & 15;
    const int row0 = blockIdx.x * MT;
    const int n0   = wave * ODIM;          // capsule column base

    // ---- async stage of one K chunk (9 async b128 ops per thread) ----
    auto issue_chunk = [&](int c) {
        const int buf = c & 1;
        // W chunk: 32 rows x 512 cols = 4096 float4; 8 per thread
        #pragma unroll
        for (int i = 0; i < 8; ++i) {
            const int idx = i * 512 + tid;              // float4 index
            const int k  = idx >> 7;                    // 0..31
            const int n4 = idx & 127;                   // float4 col
            unsigned lds = (unsigned)(uintptr_t)&ws[buf][k * WPITCH + n4 * 4];
            unsigned long long g = (unsigned long long)(uintptr_t)
                (W + (size_t)(c * KC + k) * 512 + n4 * 4);
            asm volatile("global_load_async_to_lds_b128 %0, %1, off"
                         :: "v"(lds), "v"(g) : "memory");
        }
        // x chunk: 64 rows x KC cols = 512 float4; 1 per thread
        {
            const int r = tid >> 3;                     // 0..63
            const int j = tid & 7;                      // float4 within row
            unsigned lds = (unsigned)(uintptr_t)&xs[buf][r * XPITCH + j * 4];
            unsigned long long g = (unsigned long long)(uintptr_t)
                (x + (size_t)(row0 + r) * IN_DIM + c * KC + j * 4);
            asm volatile("global_load_async_to_lds_b128 %0, %1, off"
                         :: "v"(lds), "v"(g) : "memory");
        }
    };

    v8f acc[8];                  // [mtile 0..3][ntile 0..1]
    #pragma unroll
    for (int a = 0; a < 8; ++a) acc[a] = v8f{};

    issue_chunk(0);

    for (int c = 0; c < NCHUNK; ++c) {
        if (c + 1 < NCHUNK) {
            issue_chunk(c + 1);
            asm volatile("s_wait_asynccnt 9" ::: "memory");  // chunk c landed (in-order)
        } else {
            asm volatile("s_wait_asynccnt 0" ::: "memory");
        }
        __syncthreads();

        const float* __restrict__ wsb = ws[c & 1];
        const float* __restrict__ xsb = xs[c & 1];

        #pragma unroll
        for (int ks = 0; ks < KC; ks += 4) {
            const int kk = ks + 2 * half;   // lanes 0-15: K=k,k+1 ; lanes 16-31: K=k+2,k+3
            v2f b0, b1;                     // B fragments 4x16, N = ln
            b0.x = wsb[(kk    ) * WPITCH + n0 + ln];
            b0.y = wsb[(kk + 1) * WPITCH + n0 + ln];
            b1.x = wsb[(kk    ) * WPITCH + n0 + 16 + ln];
            b1.y = wsb[(kk + 1) * WPITCH + n0 + 16 + ln];
            #pragma unroll
            for (int mt = 0; mt < 4; ++mt) {
                v2f a = *(const v2f*)&xsb[(mt * 16 + ln) * XPITCH + kk];
                acc[mt * 2 + 0] = __builtin_amdgcn_wmma_f32_16x16x4_f32(
                    false, a, false, b0, (short)0, acc[mt * 2 + 0], false, false);
                acc[mt * 2 + 1] = __builtin_amdgcn_wmma_f32_16x16x4_f32(
                    false, a, false, b1, (short)0, acc[mt * 2 + 1], false, false);
            }
        }
        __syncthreads();   // buf may be refilled by next iteration's async issue
    }

    // ---- routing in 2 passes of 32 rows; u transposed through reused ws ----
    float* ur = &ws[0][0];                 // 32 x UPITCH floats = 66 KB region

    #pragma unroll
    for (int p = 0; p < 2; ++p) {
        __syncthreads();
        #pragma unroll
        for (int t = 0; t < 2; ++t) {      // M-tiles 2p, 2p+1 -> rows-in-pass 0..31
            const int mt = p * 2 + t;
            #pragma unroll
            for (int j = 0; j < 8; ++j) {
                const int rl = t * 16 + half * 8 + j;   // D layout: lanes16-31 -> M=8+j
                ur[rl * UPITCH + n0 + ln]      = acc[mt * 2 + 0][j];
                ur[rl * UPITCH + n0 + 16 + ln] = acc[mt * 2 + 1][j];
            }
        }
        __syncthreads();

        #pragma unroll
        for (int rr = 0; rr < 2; ++rr) {
            const int rl = rr * 16 + wave;              // row within pass, wave owns it
            const int d  = lane;                        // output dim

            float u[NCAPS];
            #pragma unroll
            for (int i = 0; i < NCAPS; ++i) u[i] = ur[rl * UPITCH + i * ODIM + d];

            float b[NCAPS];
            #pragma unroll
            for (int i = 0; i < NCAPS; ++i) b[i] = 0.0f;

            float v = 0.0f;
            #pragma unroll
            for (int it = 0; it < 3; ++it) {
                float mx = b[0];
                #pragma unroll
                for (int i = 1; i < NCAPS; ++i) mx = fmaxf(mx, b[i]);
                float cexp[NCAPS];
                float csum = 0.0f;
                #pragma unroll
                for (int i = 0; i < NCAPS; ++i) { cexp[i] = expf(b[i] - mx); csum += cexp[i]; }
                const float cinv = 1.0f / csum;

                float s = 0.0f;
                #pragma unroll
                for (int i = 0; i < NCAPS; ++i) s += cexp[i] * cinv * u[i];

                float nrm = s * s;                      // wave32 butterfly over 32 dims
                #pragma unroll
                for (int off = 16; off >= 1; off >>= 1) nrm += __shfl_xor(nrm, off, 32);
                const float scale = nrm / (1.0f + nrm);
                v = scale * s / (sqrtf(nrm) + EPSV);

                #pragma unroll
                for (int i = 0; i < NCAPS; ++i) {
                    float pv = u[i] * v;
                    #pragma unroll
                    for (int off = 16; off >= 1; off >>= 1) pv += __shfl_xor(pv, off, 32);
                    b[i] += pv;
                }
            }
            out[(size_t)(row0 + p * 32 + rl) * ODIM + d] = v;
        }
    }
}

extern "C" void kernel_launch(void* const* d_in, const int* in_sizes, int n_in,
                              void* d_out, int out_size, void* d_ws, size_t ws_size,
                              hipStream_t stream) {
    const float* x = (const float*)d_in[0];   // [32768, 512]
    const float* W = (const float*)d_in[1];   // [512, 512]
    float* out = (float*)d_out;               // [32768, 32]

    const int batch = in_sizes[0] / IN_DIM;   // 32768
    const int grid  = batch / MT;             // 512 workgroups
    caps_fused_kernel<<<grid, 512, 0, stream>>>(x, W, out);
}